// Upsample_910533067305
// MI455X (gfx1250) — compile-verified
//
#include <hip/hip_runtime.h>
#include <stdint.h>

// Problem constants (from the reference): x is (16,128,128,128) f32, out is (16,128,256,256) f32.
#define H   128
#define W   128
#define OH  256
#define OW  256
#define TILE 32          // input tile edge handled per block
#define TIN  34          // input tile + 1-px halo each side
#define LP   40          // LDS row pitch (floats): halo-L at col 3, interior 4..35, halo-R at 36
#define NT  256          // threads per block (8 waves on wave32)

// ---- CDNA5 async global->LDS path (probe via __has_builtin) ----
#if defined(__has_builtin)
#  if __has_builtin(__builtin_amdgcn_global_load_async_to_lds_b32) && \
      __has_builtin(__builtin_amdgcn_s_wait_asynccnt)
#    define USE_ASYNC_LDS 1
#  endif
#  if __has_builtin(__builtin_amdgcn_global_load_async_to_lds_b128)
#    define USE_ASYNC_B128 1
#  endif
#endif
#ifndef USE_ASYNC_LDS
#  define USE_ASYNC_LDS 0
#endif
#if !USE_ASYNC_LDS
#  undef  USE_ASYNC_B128
#endif
#ifndef USE_ASYNC_B128
#  define USE_ASYNC_B128 0
#endif

typedef __attribute__((vector_size(16))) int int4v;       // b128 payload type
typedef __attribute__((address_space(1))) int   as1_int;  // "__device__ int *"
typedef __attribute__((address_space(3))) int   as3_int;  // "__shared__ int *"
typedef __attribute__((address_space(1))) int4v as1_int4; // "__device__ int4v *"
typedef __attribute__((address_space(3))) int4v as3_int4; // "__shared__ int4v *"

__global__ __launch_bounds__(NT) void upsample2x_fir_kernel(
    const float* __restrict__ x,    // (planes, H, W)
    const float* __restrict__ kw,   // 4x4 FIR kernel, row-major
    float* __restrict__ out)        // (planes, OH, OW)
{
    // LDS col (3+j) holds input col (tx0-1+j), j = 0..33. Interior cols 4..35
    // start at byte offset (r*40+4)*4 -> 16B aligned for b128 async DMA.
    __shared__ __align__(16) float tile[TIN * LP];

    const int tid   = threadIdx.x;
    const int plane = blockIdx.y;                 // b*C + c
    const int tx0   = (blockIdx.x & 3) * TILE;    // input tile origin x
    const int ty0   = (blockIdx.x >> 2) * TILE;   // input tile origin y

    const float* __restrict__ xp = x + (size_t)plane * (H * W);

    // ---- Stage 34x34 input tile (zero-padded halo) into LDS ----
    // Async DMA (ASYNCcnt) for in-range data; ds_store zeros (DScnt) for the
    // out-of-range halo. The two paths touch disjoint LDS words, so the
    // separate counters never race on the same address.
#if USE_ASYNC_B128
    // Interior: 34 rows x 8 chunks of 4 floats = 272 b128 tasks.
#pragma unroll
    for (int k = 0; k < 2; ++k) {
        const int task = tid + k * NT;
        if (task < TIN * 8) {
            const int r  = task >> 3;
            const int c  = task & 7;
            const int gy = ty0 - 1 + r;
            float* lp = &tile[r * LP + 4 + c * 4];
            if ((unsigned)gy < (unsigned)H) {
                __builtin_amdgcn_global_load_async_to_lds_b128(
                    (as1_int4*)(xp + (size_t)gy * W + tx0 + c * 4),
                    (as3_int4*)lp, /*offset=*/0, /*cpol=*/0);
            } else {
                *(float4*)lp = make_float4(0.f, 0.f, 0.f, 0.f);
            }
        }
    }
    // Halos: 34 rows x 2 columns = 68 b32 tasks.
    if (tid < TIN * 2) {
        const int r    = tid >> 1;
        const int side = tid & 1;
        const int gy   = ty0 - 1 + r;
        const int gx   = side ? (tx0 + TILE) : (tx0 - 1);
        float* lp = &tile[r * LP + (side ? 36 : 3)];
        if ((unsigned)gy < (unsigned)H && (unsigned)gx < (unsigned)W) {
            __builtin_amdgcn_global_load_async_to_lds_b32(
                (as1_int*)(xp + (size_t)gy * W + gx),
                (as3_int*)lp, /*offset=*/0, /*cpol=*/0);
        } else {
            *lp = 0.0f;
        }
    }
#else
    // Element-wise staging (b32 async if available, else plain loads).
#pragma unroll
    for (int k = 0; k < 5; ++k) {
        const int e = tid + k * NT;
        if (e < TIN * TIN) {
            const int r  = e / TIN;
            const int c  = e - r * TIN;
            const int gy = ty0 - 1 + r;
            const int gx = tx0 - 1 + c;
            float* lp = &tile[r * LP + 3 + c];
            if ((unsigned)gy < (unsigned)H && (unsigned)gx < (unsigned)W) {
#if USE_ASYNC_LDS
                __builtin_amdgcn_global_load_async_to_lds_b32(
                    (as1_int*)(xp + (size_t)gy * W + gx),
                    (as3_int*)lp, /*offset=*/0, /*cpol=*/0);
#else
                *lp = xp[(size_t)gy * W + gx];
#endif
            } else {
                *lp = 0.0f;
            }
        }
    }
#endif
#if USE_ASYNC_LDS
    __builtin_amdgcn_s_wait_asynccnt(0);
#endif
    __syncthreads();

    // ---- 4x4 kernel taps (uniform address -> scalar loads) ----
    float kv[16];
#pragma unroll
    for (int i = 0; i < 16; ++i) kv[i] = kw[i];

    float* __restrict__ op = out + (size_t)plane * (OH * OW);

    // Each thread produces a 2x2 output quad per iteration from a 3x3 LDS
    // neighborhood. Derivation (up=2, pad=(2,1), true convolution):
    //   out[2t]   = k[3]*x[t-1] + k[1]*x[t]
    //   out[2t+1] = k[2]*x[t]   + k[0]*x[t+1]      (per axis, separably)
    const int ix  = tid & 31;        // input cell x within tile: 0..31
    const int iy0 = tid >> 5;        // 0..7

#pragma unroll
    for (int it = 0; it < 4; ++it) {
        const int iy = iy0 + it * 8;                    // 0..31
        // input (iy-1, ix-1) lives at LDS [row iy][col 3+ix]
        const float* Lm = &tile[iy * LP + 3 + ix];      // input row iy-1
        const float* Lc = Lm + LP;                      // input row iy
        const float* Lp = Lc + LP;                      // input row iy+1
        const float a00 = Lm[0], a01 = Lm[1], a02 = Lm[2];
        const float a10 = Lc[0], a11 = Lc[1], a12 = Lc[2];
        const float a20 = Lp[0], a21 = Lp[1], a22 = Lp[2];

        // out(2iy  ,2ix  ) = k33*a00 + k31*a01 + k13*a10 + k11*a11
        const float o00 = kv[15]*a00 + kv[13]*a01 + kv[7]*a10 + kv[5]*a11;
        // out(2iy  ,2ix+1) = k32*a01 + k30*a02 + k12*a11 + k10*a12
        const float o01 = kv[14]*a01 + kv[12]*a02 + kv[6]*a11 + kv[4]*a12;
        // out(2iy+1,2ix  ) = k23*a10 + k21*a11 + k03*a20 + k01*a21
        const float o10 = kv[11]*a10 + kv[9]*a11 + kv[3]*a20 + kv[1]*a21;
        // out(2iy+1,2ix+1) = k22*a11 + k20*a12 + k02*a21 + k00*a22
        const float o11 = kv[10]*a11 + kv[8]*a12 + kv[2]*a21 + kv[0]*a22;

        const int oy = (ty0 + iy) * 2;
        const int ox = (tx0 + ix) * 2;
        // Coalesced: wave32 * float2 = 256 contiguous bytes per output row.
        *(float2*)(op + (size_t)oy * OW + ox)       = make_float2(o00, o01);
        *(float2*)(op + (size_t)(oy + 1) * OW + ox) = make_float2(o10, o11);
    }
}

extern "C" void kernel_launch(void* const* d_in, const int* in_sizes, int n_in,
                              void* d_out, int out_size, void* d_ws, size_t ws_size,
                              hipStream_t stream) {
    const float* x  = (const float*)d_in[0];
    const float* kw = (const float*)d_in[1];
    float* out      = (float*)d_out;

    const int planes = in_sizes[0] / (H * W);   // 16*128 = 2048
    dim3 grid((H / TILE) * (W / TILE), planes); // 16 tiles/plane, 2048 planes
    upsample2x_fir_kernel<<<grid, NT, 0, stream>>>(x, kw, out);
}